// InvariantPolynomial_63324997812583
// MI455X (gfx1250) — compile-verified
//
#include <hip/hip_runtime.h>
#include <stdint.h>

typedef __attribute__((ext_vector_type(2))) float v2f;
typedef __attribute__((ext_vector_type(8))) float v8f;

#define N_NODES  20000
#define N_GRAPHS 5000
#define N_EDGES  320000

// ---- workspace layout (bytes, 16-aligned) ----
#define OFF_W3J   0            // 27 combos * 125 floats
#define OFF_NNZC  13504        // int count
#define OFF_NNZI  13520        // 1024 uint
#define OFF_NNZV  17616        // 1024 float
#define OFF_P1    21712        // 16*432 floats (per-class FCTP1 weights)
#define OFF_TBL   49360        // 368 uint2 (output->path tables)
#define OFF_WMT   52304        // 112*144 floats (FCTP2 WmatT, col-major-ish)
#define OFF_SH    116816       // E*9 floats
#define OFF_ZZ    11636816     // E ints
#define OFF_NODE0 12916816     // N*9 floats
#define OFF_NODE1 13636816     // N*368 floats
// total = 43,076,816 bytes

// ---- FCTP1 path metadata (15 paths of IR_SH x IR_ATTR -> IR_MID) ----
__device__ const int PL1[15]  ={0,0,0,1,1,1,1,1,1,2,2,2,2,2,2};
__device__ const int PL2[15]  ={0,1,2,0,1,1,1,2,2,0,1,1,2,2,2};
__device__ const int PL3[15]  ={0,1,2,1,0,1,2,1,2,2,1,2,0,1,2};
__device__ const int PCO[15]  ={0,2,3,2,0,1,3,2,4,3,2,4,0,1,3};
__device__ const int PM3[15]  ={64,24,16,24,64,24,16,24,16,16,24,16,64,24,16};
__device__ const int PWOFF[15]={0,1024,1408,1664,2048,3072,3456,3712,4096,4352,4608,4992,5248,6272,6656};
__device__ const int PBASE[15]={0,64,88,104,128,192,216,232,256,272,288,312,328,392,416};
__device__ const int PTBASE[15]={0,1,4,9,12,13,16,21,24,29,34,39,44,45,48};
__device__ const int CPATH[5][4]={{0,4,12,-1},{5,13,-1,-1},{1,3,7,10},{2,6,9,14},{8,11,-1,-1}};
__device__ const int CNP[5]={3,2,4,4,2};

// ================= prep: build w3j tensors + weight tables on device =================
__device__ double dfact(int n){ double r=1.0; for(int i=2;i<=n;++i) r*=(double)i; return r; }

__device__ double cgc(int j1,int j2,int j3,int m1,int m2,int m3){
  if (m1+m2!=m3) return 0.0;
  double pre = sqrt((2.0*j3+1.0)*dfact(j3+j1-j2)*dfact(j3-j1+j2)*dfact(j1+j2-j3)/dfact(j1+j2+j3+1));
  pre *= sqrt(dfact(j3+m3)*dfact(j3-m3)*dfact(j1-m1)*dfact(j1+m1)*dfact(j2-m2)*dfact(j2+m2));
  int k0 = max(0, max(j2-j3-m1, j1-j3+m2));
  int k1 = min(j1+j2-j3, min(j1-m1, j2+m2));
  double s=0.0;
  for(int k=k0;k<=k1;++k){
    double d = dfact(k)*dfact(j1+j2-j3-k)*dfact(j1-m1-k)*dfact(j2+m2-k)*dfact(j3-j2+m1+k)*dfact(j3-j1-m2+k);
    s += ((k&1)? -1.0:1.0)/d;
  }
  return pre*s;
}

__device__ void uent(int l,int a,int mcol,double* re,double* im){
  *re=0.0; *im=0.0;
  int mr=a-l, c=mcol-l;
  const double inv2 = 0.70710678118654752440;
  if (mr>0){ if(c==-mr) *re=inv2; else if(c==mr) *re=((mr&1)?-1.0:1.0)*inv2; }
  else if (mr==0){ if(c==0) *re=1.0; }
  else { int mm=-mr;
    if(c==mr) *im=inv2;
    else if(c==-mr) *im = -((mm&1)?-1.0:1.0)*inv2;
  }
}

__global__ __launch_bounds__(256) void k_prep(const float* __restrict__ w1,
                                              const float* __restrict__ w2,
                                              char* __restrict__ ws){
  int tid = threadIdx.x;
  float* W3 = (float*)(ws + OFF_W3J);

  // Phase A: dense real-basis w3j tensors for all (l1,l2,l3) in [0,3)^3
  if (tid < 27){
    int l1=tid/9, l2=(tid/3)%3, l3=tid%3;
    int n1=2*l1+1, n2=2*l2+1, n3=2*l3+1;
    float* dst = W3 + tid*125;
    bool valid = (l3 >= abs(l1-l2)) && (l3 <= l1+l2);
    if(!valid){ for(int i=0;i<125;++i) dst[i]=0.f; }
    else{
      double Tre[125], Tim[125];
      int cnt=n1*n2*n3;
      for(int a=0;a<n1;++a)for(int b=0;b<n2;++b)for(int c=0;c<n3;++c){
        double sre=0.0, sim=0.0;
        for(int m=-l1;m<=l1;++m)for(int n=-l2;n<=l2;++n){
          int o=m+n; if(o<-l3||o>l3) continue;
          double cg=cgc(l1,l2,l3,m,n,o); if(cg==0.0) continue;
          double u1r,u1i,u2r,u2i,u3r,u3i;
          uent(l1,a,m+l1,&u1r,&u1i);
          uent(l2,b,n+l2,&u2r,&u2i);
          uent(l3,c,o+l3,&u3r,&u3i);
          u3i = -u3i; // conj
          double pr = u1r*u2r - u1i*u2i, pi = u1r*u2i + u1i*u2r;
          double qr = pr*u3r - pi*u3i,   qi = pr*u3i + pi*u3r;
          sre += qr*cg; sim += qi*cg;
        }
        Tre[(a*n2+b)*n3+c]=sre; Tim[(a*n2+b)*n3+c]=sim;
      }
      double mr=0.0,mi=0.0;
      for(int i=0;i<cnt;++i){ mr=fmax(mr,fabs(Tre[i])); mi=fmax(mi,fabs(Tim[i])); }
      double nrm=0.0;
      if (mr>=mi){ for(int i=0;i<cnt;++i) nrm+=Tre[i]*Tre[i]; nrm=sqrt(nrm);
                   for(int i=0;i<cnt;++i) dst[i]=(float)(Tre[i]/nrm); }
      else       { for(int i=0;i<cnt;++i) nrm+=Tim[i]*Tim[i]; nrm=sqrt(nrm);
                   for(int i=0;i<cnt;++i) dst[i]=(float)(Tim[i]/nrm); }
      for(int i=cnt;i<125;++i) dst[i]=0.f;
    }
  }
  __syncthreads();

  // Phase B: sparse nnz list for the 15 FCTP1 w3j contractions (thread 0)
  if (tid==0){
    unsigned* NI=(unsigned*)(ws+OFF_NNZI);
    float*    NV=(float*)(ws+OFF_NNZV);
    const int LB[3]={0,1,4};
    int cnt=0;
    for(int p=0;p<15;++p){
      int l1=PL1[p], l2=PL2[p], l3=PL3[p];
      int n1=2*l1+1, n2=2*l2+1, n3=2*l3+1;
      const float* C = W3 + ((l1*3+l2)*3+l3)*125;
      for(int i=0;i<n1;++i)for(int j=0;j<n2;++j)for(int k=0;k<n3;++k){
        float v=C[(i*n2+j)*n3+k];
        if (fabsf(v)>1e-7f){
          NI[cnt]=(unsigned)((LB[l1]+i) | ((LB[l2]+j)<<4) | ((PTBASE[p]+k)<<8));
          NV[cnt]=v; ++cnt;
        }
      }
    }
    *(int*)(ws+OFF_NNZC)=cnt;
  }

  // Phase C: P1 per-class FCTP1 weights: alpha_c * 4(one-hot scale) * inv_nn(node0)
  {
    float* P1=(float*)(ws+OFF_P1);
    const double ALPHAC[5]={ sqrt(1.0/48.0), sqrt(3.0/32.0), sqrt(3.0/64.0),
                             sqrt(5.0/64.0), sqrt(5.0/32.0) };
    const double invnn = 1.0/sqrt(3.0);
    for(int idx=tid; idx<6912; idx+=256){
      int cls=idx/432, r=idx%432;
      int p=0;
      for(int q=14;q>=0;--q){ if(r>=PBASE[q]){ p=q; break; } }
      int w=r-PBASE[p];
      P1[idx]=(float)(ALPHAC[PCO[p]]*4.0*invnn*(double)w1[PWOFF[p]+cls*PM3[p]+w]);
    }
  }

  // Phase D: per-output (368) path tables, 4x 16-bit fields: (p1idx | tidx<<9)
  {
    uint2* TB=(uint2*)(ws+OFF_TBL);
    for(int o=tid;o<368;o+=256){
      int c,start,d;
      if(o<64){c=0;start=0;d=1;} else if(o<136){c=1;start=64;d=3;}
      else if(o<208){c=2;start=136;d=3;} else if(o<288){c=3;start=208;d=5;}
      else {c=4;start=288;d=5;}
      int rel=o-start, w=rel/d, k=rel%d;
      unsigned f[4]={0,0,0,0};
      for(int j=0;j<CNP[c];++j){
        int p=CPATH[c][j];
        f[j]=(unsigned)((PBASE[p]+w) | ((PTBASE[p]+k)<<9));
      }
      TB[o]=make_uint2(f[0]|(f[1]<<16), f[2]|(f[3]<<16));
    }
  }

  // Phase E: FCTP2 combined weights WmatT[n=112][k=144], all scales folded:
  // alpha_{c3} * 4 * w3j(l,l,0)diag * inv_nn(node1) * inv_nn(node2) * 0.5
  {
    float* WT=(float*)(ws+OFF_WMT);
    const double a0=1.0/sqrt(640.0), a1=1.0/sqrt(1664.0);
    const double g2=4.0*(1.0/3.0)*0.5;
    double s000=(double)W3[((0*3+0)*3+0)*125 + 0];
    double s110=(double)W3[((1*3+1)*3+0)*125 + 0];
    double s220=(double)W3[((2*3+2)*3+0)*125 + 0];
    for(int idx=tid; idx<16128; idx+=256){
      int n=idx/144, u=idx%144;
      int c=n/7, w=n%7;
      double val=0.0;
      if (w==0){ // output co0 = (1,0,-)
        if (u>=64 && u<88)        val=a0*g2*s110*(double)w2[6144 +((u-64 )*16+c)];
        else if (u>=128)          val=a0*g2*s220*(double)w2[10368+((u-128)*16+c)];
      } else {   // output co1 = (6,0,+), w-1 in [0,6)
        int wo=w-1;
        if (u<64)                 val=a1*g2*s000*(double)w2[        (u      *16+c)*6+wo];
        else if (u>=88 && u<112)  val=a1*g2*s110*(double)w2[6528 +((u-88 )*16+c)*6+wo];
        else if (u>=112 && u<128) val=a1*g2*s220*(double)w2[8832 +((u-112)*16+c)*6+wo];
      }
      WT[n*144+u]=(float)val;
    }
  }
}

// ================= K0: edge geometry, sh, zz, node0 = segsum(sh, dst) =================
__global__ __launch_bounds__(256) void k_edge(const float* __restrict__ pos,
                                              const int* __restrict__ zi,
                                              const int* __restrict__ esrc,
                                              const int* __restrict__ edst,
                                              float* __restrict__ shb,
                                              int* __restrict__ zzb,
                                              float* __restrict__ node0){
  int e = blockIdx.x*256 + threadIdx.x;
  if (e >= N_EDGES) return;
  int s=esrc[e], d=edst[e];
  float x=pos[3*s]-pos[3*d], y=pos[3*s+1]-pos[3*d+1], z=pos[3*s+2]-pos[3*d+2];
  float r2=x*x+y*y+z*z;
  const float s3=1.7320508075688772f, s15=3.8729833462074170f, s5=2.2360679774997897f;
  float v[9];
  v[0]=1.f; v[1]=s3*y; v[2]=s3*z; v[3]=s3*x;
  v[4]=s15*x*y; v[5]=s15*y*z; v[6]=0.5f*s5*(3.f*z*z-r2); v[7]=s15*x*z; v[8]=0.5f*s15*(x*x-y*y);
  #pragma unroll
  for(int i=0;i<9;++i){ shb[e*9+i]=v[i]; atomicAdd(&node0[d*9+i], v[i]); }
  zzb[e]=4*zi[s]+zi[d];
}

// ================= K1: FCTP1 per edge (wave/edge) + atomic scatter to node1 =================
__global__ __launch_bounds__(256) void k_fctp1(const float* __restrict__ shb,
                                               const int* __restrict__ zzb,
                                               const int* __restrict__ esrc,
                                               const int* __restrict__ edst,
                                               const float* __restrict__ node0,
                                               float* __restrict__ node1,
                                               const float* __restrict__ gP1,
                                               const uint2* __restrict__ gTBL,
                                               const unsigned* __restrict__ gNI,
                                               const float* __restrict__ gNV,
                                               const int* __restrict__ gCnt){
  __shared__ float sP1[6912];
  __shared__ uint2 sTBL[368];
  __shared__ unsigned sNI[768];
  __shared__ float sNV[768];
  __shared__ float sT[8][80];  // [wave]: [0..8]=node0[src], [9..17]=sh, [20..72]=t(53)

  int tid=threadIdx.x;
  int nnzCnt = gCnt[0];
  for(int i=tid;i<6912;i+=256) sP1[i]=gP1[i];
  for(int i=tid;i<368;i+=256)  sTBL[i]=gTBL[i];
  for(int i=tid;i<nnzCnt;i+=256){ sNI[i]=gNI[i]; sNV[i]=gNV[i]; }
  __syncthreads();

  int wv=tid>>5, lane=tid&31;
  float* TX=&sT[wv][0];

  for(int it=0; it<10; ++it){
    int e=(blockIdx.x*8+wv)*10+it;
    int src=esrc[e], dst=edst[e];
    if (lane<9)        TX[lane]=node0[src*9+lane];
    else if (lane<18)  TX[lane]=shb[e*9+(lane-9)];
    TX[20+lane]=0.f;
    if (lane<21) TX[52+lane]=0.f;
    asm volatile("s_wait_dscnt 0x0" ::: "memory");

    for(int n=lane;n<nnzCnt;n+=32){
      unsigned pi=sNI[n];
      float prod = sNV[n]*TX[pi&15]*TX[9+((pi>>4)&15)];
      atomicAdd(&TX[20+(pi>>8)], prod);
    }
    asm volatile("s_wait_dscnt 0x0" ::: "memory");

    int cls432 = zzb[e]*432;
    for(int o=lane;o<368;o+=32){
      int np;
      if(o<64) np=3; else if(o<136) np=2; else if(o<288) np=4; else np=2;
      uint2 f=sTBL[o];
      unsigned f0=f.x&0xFFFFu, f1=f.x>>16, f2=f.y&0xFFFFu, f3=f.y>>16;
      float acc = sP1[cls432+(f0&511u)]*TX[20+(f0>>9)];
      if(np>1) acc += sP1[cls432+(f1&511u)]*TX[20+(f1>>9)];
      if(np>2) acc += sP1[cls432+(f2&511u)]*TX[20+(f2>>9)];
      if(np>3) acc += sP1[cls432+(f3&511u)]*TX[20+(f3>>9)];
      atomicAdd(&node1[dst*368+o], acc);
    }
    asm volatile("s_wait_dscnt 0x0" ::: "memory");
  }
}

// ================= K2: FCTP2 via WMMA (16 edges x 144) @ (144 x 112 classes*7) =================
__global__ __launch_bounds__(128) void k_fctp2(const float* __restrict__ shb,
                                               const int* __restrict__ zzb,
                                               const int* __restrict__ esrc,
                                               const int* __restrict__ edst,
                                               const int* __restrict__ batch,
                                               const float* __restrict__ node1,
                                               const float* __restrict__ gWT,
                                               float* __restrict__ out){
  __shared__ float sW[16128];       // WmatT [n=112][k=144]
  __shared__ int sZZ[4][16];
  __shared__ int sBD[4][16];

  int tid=threadIdx.x;
  for(int i=tid;i<16128;i+=128) sW[i]=gWT[i];
  __syncthreads();

  int wv=tid>>5, lane=tid&31;
  int col=lane&15;
  int off=(lane<16)?0:2;

  for(int tt=0;tt<4;++tt){
    int tile=(blockIdx.x*4+wv)*4+tt;
    int e=tile*16+col;
    int src=esrc[e];
    const float* n1 = node1 + src*368;
    float shv[9];
    #pragma unroll
    for(int i=0;i<9;++i) shv[i]=shb[e*9+i];
    if (lane<16){ sZZ[wv][lane]=zzb[e]; sBD[wv][lane]=batch[edst[e]]; }
    asm volatile("s_wait_dscnt 0x0" ::: "memory");

    // A fragments: lane row = col, K parity half via off. ar[2s],ar[2s+1] = A[row][4s+off+{0,1}]
    float ar[72];
    #pragma unroll
    for(int s=0;s<36;++s){
      const int ub=4*s;
      int u0=ub+off;
      float a0,a1;
      if (ub<64){ a0=n1[u0]; a1=n1[u0+1]; }
      else if (ub<88){
        const float* p=n1+64+(u0-64)*3;
        a0=p[0]*shv[1]+p[1]*shv[2]+p[2]*shv[3];
        a1=p[3]*shv[1]+p[4]*shv[2]+p[5]*shv[3];
      } else if (ub<112){
        const float* p=n1+136+(u0-88)*3;
        a0=p[0]*shv[1]+p[1]*shv[2]+p[2]*shv[3];
        a1=p[3]*shv[1]+p[4]*shv[2]+p[5]*shv[3];
      } else if (ub<128){
        const float* p=n1+208+(u0-112)*5;
        a0=p[0]*shv[4]+p[1]*shv[5]+p[2]*shv[6]+p[3]*shv[7]+p[4]*shv[8];
        a1=p[5]*shv[4]+p[6]*shv[5]+p[7]*shv[6]+p[8]*shv[7]+p[9]*shv[8];
      } else {
        const float* p=n1+288+(u0-128)*5;
        a0=p[0]*shv[4]+p[1]*shv[5]+p[2]*shv[6]+p[3]*shv[7]+p[4]*shv[8];
        a1=p[5]*shv[4]+p[6]*shv[5]+p[7]*shv[6]+p[8]*shv[7]+p[9]*shv[8];
      }
      ar[2*s]=a0; ar[2*s+1]=a1;
    }

    for(int nt=0;nt<7;++nt){
      v8f acc = {};
      int bbase=(nt*16+col)*144+off;
      #pragma unroll
      for(int s=0;s<36;++s){
        v2f a; a.x=ar[2*s]; a.y=ar[2*s+1];
        v2f b; b.x=sW[bbase+4*s]; b.y=sW[bbase+4*s+1];
        acc = __builtin_amdgcn_wmma_f32_16x16x4_f32(false, a, false, b, (short)0, acc, false, false);
      }
      // C layout: VGPR r holds rows M=r (lanes 0-15) / M=r+8 (lanes 16-31), col N=lane&15
      #pragma unroll
      for(int r8=0;r8<8;++r8){
        int M=r8+((lane<16)?0:8);
        int n=nt*16+col;
        int zb=sZZ[wv][M]*7;
        int dn=n-zb;
        if (dn>=0 && dn<7) atomicAdd(&out[sBD[wv][M]*7+dn], acc[r8]);
      }
    }
  }
}

// ================= zero helper =================
__global__ void k_zero4(float4* __restrict__ p, int n4){
  int i=blockIdx.x*256+threadIdx.x;
  if (i<n4) p[i]=make_float4(0.f,0.f,0.f,0.f);
}

extern "C" void kernel_launch(void* const* d_in, const int* in_sizes, int n_in,
                              void* d_out, int out_size, void* d_ws, size_t ws_size,
                              hipStream_t stream){
  const float* pos  =(const float*)d_in[0];
  const int*   zi   =(const int*)d_in[1];
  const int*   batch=(const int*)d_in[2];
  const int*   esrc =(const int*)d_in[3];
  const int*   edst =(const int*)d_in[4];
  const float* w1   =(const float*)d_in[5];
  const float* w2   =(const float*)d_in[6];
  char* ws=(char*)d_ws;
  float* out=(float*)d_out;

  // zero node0+node1 (contiguous) and the output accumulator
  k_zero4<<<(1885000+255)/256,256,0,stream>>>((float4*)(ws+OFF_NODE0), 1885000);
  k_zero4<<<(8750+255)/256,   256,0,stream>>>((float4*)out, 8750);

  k_prep<<<1,256,0,stream>>>(w1, w2, ws);

  k_edge<<<N_EDGES/256,256,0,stream>>>(pos, zi, esrc, edst,
                                       (float*)(ws+OFF_SH), (int*)(ws+OFF_ZZ),
                                       (float*)(ws+OFF_NODE0));

  k_fctp1<<<4000,256,0,stream>>>((const float*)(ws+OFF_SH), (const int*)(ws+OFF_ZZ),
                                 esrc, edst,
                                 (const float*)(ws+OFF_NODE0), (float*)(ws+OFF_NODE1),
                                 (const float*)(ws+OFF_P1), (const uint2*)(ws+OFF_TBL),
                                 (const unsigned*)(ws+OFF_NNZI), (const float*)(ws+OFF_NNZV),
                                 (const int*)(ws+OFF_NNZC));

  k_fctp2<<<1250,128,0,stream>>>((const float*)(ws+OFF_SH), (const int*)(ws+OFF_ZZ),
                                 esrc, edst, batch,
                                 (const float*)(ws+OFF_NODE1),
                                 (const float*)(ws+OFF_WMT), out);
}